// GHMC_Loss_7164005449994
// MI455X (gfx1250) — compile-verified
//
#include <hip/hip_runtime.h>

typedef __attribute__((ext_vector_type(2))) float v2f;
typedef __attribute__((ext_vector_type(8))) float v8f;

#define NCLS 512          // C
#define WPB  8            // waves per block in pass1

// ---------------------------------------------------------------- pass 0
__global__ void ghm_init(unsigned* __restrict__ hist) {
    if (threadIdx.x < 16) hist[threadIdx.x] = 0u;
}

// ---------------------------------------------------------------- pass 1
// One wave32 per row. Streams pred once: softmax stats, CE, per-row 10-bin
// counts, global histogram via LDS aggregation.
__global__ void __launch_bounds__(256) ghm_pass1(
    const float* __restrict__ pred, const int* __restrict__ target,
    unsigned* __restrict__ hist, float* __restrict__ ce,
    float* __restrict__ rowcnt, int nrows)
{
    __shared__ unsigned lhist[10];
    const int tid  = threadIdx.x;
    if (tid < 10) lhist[tid] = 0u;
    __syncthreads();

    const int wave = tid >> 5;
    const int lane = tid & 31;
    const int row  = blockIdx.x * WPB + wave;

    if (row < nrows) {
        const float4* rp = (const float4*)(pred + (size_t)row * NCLS);
        float x[16];
        #pragma unroll
        for (int i = 0; i < 4; ++i) {
            float4 v = rp[i * 32 + lane];
            x[i * 4 + 0] = v.x; x[i * 4 + 1] = v.y;
            x[i * 4 + 2] = v.z; x[i * 4 + 3] = v.w;
        }

        // row max
        float m = x[0];
        #pragma unroll
        for (int j = 1; j < 16; ++j) m = fmaxf(m, x[j]);
        #pragma unroll
        for (int off = 16; off >= 1; off >>= 1) m = fmaxf(m, __shfl_xor(m, off));

        // sum of exp
        float e[16];
        float s = 0.0f;
        #pragma unroll
        for (int j = 0; j < 16; ++j) { e[j] = __expf(x[j] - m); s += e[j]; }
        #pragma unroll
        for (int off = 16; off >= 1; off >>= 1) s += __shfl_xor(s, off);
        const float inv = 1.0f / s;

        const int tgt = target[row];
        float pt = 0.0f, tsum = 0.0f;
        unsigned long long pk0 = 0ull, pk1 = 0ull;   // 5 bins x 12-bit fields each

        #pragma unroll
        for (int i = 0; i < 4; ++i) {
            #pragma unroll
            for (int c = 0; c < 4; ++c) {
                const int elem = i * 128 + lane * 4 + c;
                const float p  = e[i * 4 + c] * inv;
                tsum += __expf(p);                    // for log-softmax of p
                float g = p;
                if (elem == tgt) { pt = p; g = 1.0f - p; }
                int b = (int)(g * 10.0f);
                b = (b > 9) ? 9 : ((b < 0) ? 0 : b);
                if (b < 5) pk0 += 1ull << (b * 12);
                else       pk1 += 1ull << ((b - 5) * 12);
            }
        }
        #pragma unroll
        for (int off = 16; off >= 1; off >>= 1) {
            pt   += __shfl_xor(pt, off);
            tsum += __shfl_xor(tsum, off);
            pk0  += __shfl_xor(pk0, off);
            pk1  += __shfl_xor(pk1, off);
        }

        if (lane == 0) {
            ce[row] = __logf(tsum) - pt;
            float* rc = rowcnt + (size_t)row * 16;
            #pragma unroll
            for (int b = 0; b < 5; ++b) {
                unsigned c0 = (unsigned)((pk0 >> (b * 12)) & 0xFFFull);
                unsigned c1 = (unsigned)((pk1 >> (b * 12)) & 0xFFFull);
                rc[b]     = (float)c0;
                rc[b + 5] = (float)c1;
                if (c0) atomicAdd(&lhist[b], c0);
                if (c1) atomicAdd(&lhist[b + 5], c1);
            }
            #pragma unroll
            for (int b = 10; b < 16; ++b) rc[b] = 0.0f;   // zero pad for WMMA K
        }
    }
    __syncthreads();
    if (tid < 10 && lhist[tid]) atomicAdd(&hist[tid], lhist[tid]);
}

// ---------------------------------------------------------------- pass 2
// Bin weights w_b = (total / (max(n_valid,1) * 0.1 * count_b))^0.75, zero out.
__global__ void ghm_pass2(const unsigned* __restrict__ hist,
                          float* __restrict__ wpow, float* __restrict__ out,
                          float total)
{
    const int i = threadIdx.x;            // 32 threads
    const unsigned c = (i < 10) ? hist[i] : 0u;
    const unsigned long long ball = __ballot(i < 10 && c > 0u);
    const float nvalid = (float)__popcll(ball);

    float w = 0.0f;
    if (i < 10) {
        if (c > 0u) {
            const float acc = 0.1f * (float)c;                 // (1-momentum)*count
            const float bw  = total / (fmaxf(nvalid, 1.0f) * acc);
            w = powf(bw, 0.75f);                               // alpha
        }
        if (nvalid <= 0.0f) w = 1.0f;                          // n_valid==0 path
    }
    if (i < 16) wpow[i] = w;
    if (i == 0) out[0] = 0.0f;
}

// ---------------------------------------------------------------- pass 3
// 16 rows per wave: D = A(16x4 counts) x B(4x16 broadcast weights) via
// V_WMMA_F32_16X16X4_F32, accumulated over 3 K-chunks (bins 0..9, zero pad).
// loss_row = ce[row] * D[row][0]; mean accumulated atomically.
__global__ void __launch_bounds__(256) ghm_pass3(
    const float* __restrict__ rowcnt, const float* __restrict__ ce,
    const float* __restrict__ wpow, float* __restrict__ out,
    float scale /* 1/(C*N) */)
{
    const int tid  = threadIdx.x;
    const int wave = tid >> 5;
    const int lane = tid & 31;
    const int r0   = (blockIdx.x * 8 + wave) * 16;

    // ISA 32-bit A 16x4 layout: lanes 0-15 carry K=0 (v0) / K=1 (v1),
    // lanes 16-31 carry K=2 / K=3, M = lane & 15.
    const int mrow = r0 + (lane & 15);
    const int koff = (lane < 16) ? 0 : 2;
    const float* rc = rowcnt + (size_t)mrow * 16;

    v8f acc = {};
    #pragma unroll
    for (int k0 = 0; k0 < 12; k0 += 4) {
        v2f a, b;
        a.x = rc[k0 + koff];
        a.y = rc[k0 + koff + 1];
        b.x = wpow[k0 + koff];          // B[k][n] = w[k] broadcast over n
        b.y = wpow[k0 + koff + 1];
        acc = __builtin_amdgcn_wmma_f32_16x16x4_f32(
            /*neg_a=*/false, a, /*neg_b=*/false, b,
            /*c_mod=*/(short)0, acc, /*reuse_a=*/false, /*reuse_b=*/false);
    }

    // C/D layout: lane 0 -> (M=0..7, N=0) in acc[0..7]; lane 16 -> M=8..15.
    float part = 0.0f;
    if ((lane & 15) == 0) {
        const int base = r0 + ((lane == 16) ? 8 : 0);
        #pragma unroll
        for (int v = 0; v < 8; ++v) part += acc[v] * ce[base + v];
    }
    part += __shfl_xor(part, 16);
    if (lane == 0) atomicAdd(out, part * scale);
}

// ---------------------------------------------------------------- launch
extern "C" void kernel_launch(void* const* d_in, const int* in_sizes, int n_in,
                              void* d_out, int out_size, void* d_ws, size_t ws_size,
                              hipStream_t stream) {
    const float* pred = (const float*)d_in[0];
    const int*   tgt  = (const int*)d_in[1];
    const int nrows   = in_sizes[1];            // 131072
    float* out        = (float*)d_out;

    // Workspace layout:
    //   [0,64)                 : unsigned hist[16]
    //   [64,128)               : float wpow[16]
    //   [128, 128+4N)          : float ce[N]
    //   [128+4N, 128+4N+64N)   : float rowcnt[N*16]   (~8.9 MB total)
    unsigned char* ws = (unsigned char*)d_ws;
    unsigned* hist   = (unsigned*)ws;
    float*    wpow   = (float*)(ws + 64);
    float*    cearr  = (float*)(ws + 128);
    float*    rowcnt = (float*)(ws + 128 + (size_t)nrows * 4);

    const float total = (float)nrows * (float)NCLS;

    ghm_init <<<1, 32, 0, stream>>>(hist);
    ghm_pass1<<<(nrows + WPB - 1) / WPB, 32 * WPB, 0, stream>>>(
        pred, tgt, hist, cearr, rowcnt, nrows);
    ghm_pass2<<<1, 32, 0, stream>>>(hist, wpow, out, total);
    ghm_pass3<<<nrows / 128, 256, 0, stream>>>(
        rowcnt, cearr, wpow, out, 1.0f / total);
}